// QFCC_Model_26792005992463
// MI455X (gfx1250) — compile-verified
//
#include <hip/hip_runtime.h>
#include <hip/hip_bf16.h>

// ---------------------------------------------------------------------------
// 12-qubit state-vector simulator, fully fused for MI455X (gfx1250).
//  - encoding layer evaluated in closed form (product state)
//  - per-layer rotations fused into three 16x16 complex unitaries
//    applied with chained V_WMMA_F32_16X16X4_F32 (4 real chains / tile)
//  - CNOT ladder folded into a single Gray-code permutation k ^ (k>>1)
//  - one workgroup (8 waves) per batch element, state resident in LDS
//  - Ai negated once per group (keeps LDS->WMMA B path dependency-free)
//  - Dr/Di accumulator chains interleaved for back-to-back WMMA issue
// ---------------------------------------------------------------------------

typedef float v2f __attribute__((ext_vector_type(2)));
typedef float v8f __attribute__((ext_vector_type(8)));

#define NQ     12
#define DIM    4096     // 2^12
#define LAYERS 6

__device__ __forceinline__ v8f wmma4(v2f a, v2f b, v8f c) {
  // D(16x16,f32) += A(16x4,f32) * B(4x16,f32)
  return __builtin_amdgcn_wmma_f32_16x16x4_f32(
      /*neg_a=*/false, a, /*neg_b=*/false, b,
      /*c_mod=*/(short)0, c, /*reuse_a=*/false, /*reuse_b=*/false);
}

// ---------------------------------------------------------------------------
// Setup: build 18 fused 16x16 complex unitaries (6 layers x 3 qubit-groups).
// U[i,j] = prod_{t=0..3} R(w[l, 4g+t]) [ bit_t(i), bit_t(j) ]
// Stored in ws as [mat][0:256]=real, [mat][256:512]=imag, row-major 16x16.
// ---------------------------------------------------------------------------
__global__ void qfcc_build_unitaries(const float* __restrict__ w,
                                     float* __restrict__ U) {
  const int mat = blockIdx.x;            // 0..17
  const int l = mat / 3, g = mat % 3;
  const int t = threadIdx.x;             // 0..255
  const int i = t >> 4, j = t & 15;

  float ar = 1.f, ai = 0.f;
#pragma unroll
  for (int q4 = 0; q4 < 4; ++q4) {
    const int q = 4 * g + q4;
    const float phi = w[(l * 12 + q) * 3 + 0];
    const float th  = w[(l * 12 + q) * 3 + 1];
    const float om  = w[(l * 12 + q) * 3 + 2];
    const float c  = cosf(0.5f * th);
    const float s  = sinf(0.5f * th);
    const float al = 0.5f * (phi + om);
    const float be = 0.5f * (phi - om);
    const int it = (i >> (3 - q4)) & 1;
    const int jt = (j >> (3 - q4)) & 1;
    float rr, ri;
    if (it == 0 && jt == 0)      { rr =  cosf(al) * c; ri = -sinf(al) * c; }
    else if (it == 0)            { rr = -cosf(be) * s; ri = -sinf(be) * s; }
    else if (jt == 0)            { rr =  cosf(be) * s; ri = -sinf(be) * s; }
    else                         { rr =  cosf(al) * c; ri =  sinf(al) * c; }
    const float nr = ar * rr - ai * ri;
    const float ni = ar * ri + ai * rr;
    ar = nr; ai = ni;
  }
  U[mat * 512 +       i * 16 + j] = ar;
  U[mat * 512 + 256 + i * 16 + j] = ai;
}

// ---------------------------------------------------------------------------
// Main kernel: one block per batch element, 256 threads = 8 wave32.
// ---------------------------------------------------------------------------
__global__ __launch_bounds__(256) void qfcc_sim_kernel(
    const float* __restrict__ x,        // [1024,12]
    const float* __restrict__ U,        // [18][2][256] fused unitaries
    const float* __restrict__ head_w,   // [2,1]
    const float* __restrict__ head_b,   // [2]
    float* __restrict__ out)            // [1024,2]
{
  __shared__ float Sr[DIM];
  __shared__ float Si[DIM];
  __shared__ float red[8];
  __shared__ float cs[NQ], sn[NQ];

  const int t = threadIdx.x;
  const int b = blockIdx.x;

  // ---- encoding layer: closed-form product state --------------------------
  if (t < NQ) {
    const float h = 0.5f * x[b * NQ + t];
    cs[t] = cosf(h);
    sn[t] = sinf(h);
  }
  __syncthreads();

#pragma unroll
  for (int u = 0; u < 16; ++u) {
    const int k = t + 256 * u;
    float m = 1.f;
#pragma unroll
    for (int q = 0; q < NQ; ++q)
      m *= ((k >> (11 - q)) & 1) ? sn[q] : cs[q];
    const int pc = __popc(k) & 3;             // (-i)^popcount phase
    Sr[k] = (pc == 0) ? m : ((pc == 2) ? -m : 0.f);
    Si[k] = (pc == 1) ? -m : ((pc == 3) ? m : 0.f);
  }
  __syncthreads();

  const int wave = t >> 5;
  const int lane = t & 31;
  const int row  = lane & 15;     // M row (A/D) or N column (B)
  const int hi   = lane >> 4;     // lane half selects K/M offsets

  for (int l = 0; l < LAYERS; ++l) {
    for (int g = 0; g < 3; ++g) {
      const float* Um = U + (l * 3 + g) * 512;

      // A fragments (unitary), ISA 16x4 f32 layout:
      //   VGPR0: K=4kk+0 (lanes 0-15) / K=4kk+2 (lanes 16-31); VGPR1: +1
      // Ain = -Ai computed once per group so the complex-GEMM subtraction
      // never puts a VALU op behind the per-tile LDS B loads.
      v2f Ar[4], Ai[4], Ain[4];
#pragma unroll
      for (int kk = 0; kk < 4; ++kk) {
        const int c0 = 4 * kk + 2 * hi;
        Ar[kk].x  = Um[row * 16 + c0];
        Ar[kk].y  = Um[row * 16 + c0 + 1];
        Ai[kk].x  = Um[256 + row * 16 + c0];
        Ai[kk].y  = Um[256 + row * 16 + c0 + 1];
        Ain[kk].x = -Ai[kk].x;
        Ain[kk].y = -Ai[kk].y;
      }

      // flat index for group g: idx = (c>>s)<<(s+4) | j<<s | (c & (W-1))
      const int s   = 8 - 4 * g;
      const int Wm1 = (1 << s) - 1;

      // 16 column-tiles of 16; each wave owns 2 disjoint tiles -> no barrier
      for (int tt = 0; tt < 2; ++tt) {
        const int cN  = (wave * 2 + tt) * 16 + row;   // this lane's column
        const int chi = (cN >> s) << (s + 4);
        const int clo = cN & Wm1;

        v2f Br[4], Bi[4];
#pragma unroll
        for (int kk = 0; kk < 4; ++kk) {
          const int j0 = 4 * kk + 2 * hi;
          const int i0 = chi + (j0 << s) + clo;
          const int i1 = chi + ((j0 + 1) << s) + clo;
          Br[kk].x = Sr[i0];  Br[kk].y = Sr[i1];
          Bi[kk].x = Si[i0];  Bi[kk].y = Si[i1];
        }

        // complex GEMM: Dr = Ar*Br + (-Ai)*Bi ; Di = Ar*Bi + Ai*Br
        // Interleave the two independent accumulator chains.
        v8f Dr = {}; v8f Di = {};
#pragma unroll
        for (int kk = 0; kk < 4; ++kk) {
          Dr = wmma4(Ar[kk],  Br[kk], Dr);
          Di = wmma4(Ar[kk],  Bi[kk], Di);
        }
#pragma unroll
        for (int kk = 0; kk < 4; ++kk) {
          Dr = wmma4(Ain[kk], Bi[kk], Dr);
          Di = wmma4(Ai[kk],  Br[kk], Di);
        }

        // D layout: VGPR v holds M=v (lanes 0-15) / M=v+8 (lanes 16-31), N=row
#pragma unroll
        for (int v = 0; v < 8; ++v) {
          const int idx = chi + ((v + 8 * hi) << s) + clo;
          Sr[idx] = Dr[v];
          Si[idx] = Di[v];
        }
      }
      __syncthreads();
    }

    // ---- CNOT ladder == Gray-code permutation ----------------------------
    float tr[16], ti[16];
#pragma unroll
    for (int u = 0; u < 16; ++u) {
      const int k = t + 256 * u;
      const int src = k ^ (k >> 1);
      tr[u] = Sr[src];
      ti[u] = Si[src];
    }
    __syncthreads();
#pragma unroll
    for (int u = 0; u < 16; ++u) {
      const int k = t + 256 * u;
      Sr[k] = tr[u];
      Si[k] = ti[u];
    }
    __syncthreads();
  }

  // ---- <Z0> measurement: wave32 shuffle reduction + tiny LDS combine -----
  float acc = 0.f;
#pragma unroll
  for (int u = 0; u < 16; ++u) {
    const int k = t + 256 * u;
    const float p = Sr[k] * Sr[k] + Si[k] * Si[k];
    acc += (k & 2048) ? -p : p;     // qubit 0 is the MSB
  }
#pragma unroll
  for (int off = 16; off > 0; off >>= 1)
    acc += __shfl_down(acc, off, 32);
  if (lane == 0) red[wave] = acc;
  __syncthreads();
  if (t == 0) {
    float z = 0.f;
#pragma unroll
    for (int wv = 0; wv < 8; ++wv) z += red[wv];
    out[b * 2 + 0] = z * head_w[0] + head_b[0];
    out[b * 2 + 1] = z * head_w[1] + head_b[1];
  }
}

// ---------------------------------------------------------------------------
extern "C" void kernel_launch(void* const* d_in, const int* in_sizes, int n_in,
                              void* d_out, int out_size, void* d_ws, size_t ws_size,
                              hipStream_t stream) {
  const float* x       = (const float*)d_in[0];   // [1024,12]
  const float* weights = (const float*)d_in[1];   // [6,12,3]
  const float* head_w  = (const float*)d_in[2];   // [2,1]
  const float* head_b  = (const float*)d_in[3];   // [2]
  float*       out     = (float*)d_out;           // [1024,2]
  float*       U       = (float*)d_ws;            // 18 * 512 floats = 36 KB

  qfcc_build_unitaries<<<18, 256, 0, stream>>>(weights, U);
  qfcc_sim_kernel<<<1024, 256, 0, stream>>>(x, U, head_w, head_b, out);
}